// GraphSensorGNN_58669253263485
// MI455X (gfx1250) — compile-verified
//
#include <hip/hip_runtime.h>
#include <math.h>

// Problem constants (match reference): B=4, N=256, D=128, CD=3, SIGMA=1, 1 layer.
#define BDIM 4
#define NPTS 256
#define DDIM 128
#define CDIM 3
#define BN   (BDIM * NPTS)   // 1024 rows in all GEMMs

typedef __attribute__((ext_vector_type(2))) float v2f;
typedef __attribute__((ext_vector_type(8))) float v8f;

// ---------------------------------------------------------------------------
// One 16x16 f32 output tile via V_WMMA_F32_16X16X4_F32, K-loop in steps of 4.
// A: X row-major (ld = ldx). Lanes 0-15 hold K=+0,+1 (2 VGPRs), lanes 16-31
// hold K=+2,+3 per the CDNA5 ISA 16x4 A layout. B mirrored (N across lanes).
// ---------------------------------------------------------------------------
__device__ __forceinline__ v8f wmma_gemm_tile(const float* __restrict__ X, int ldx,
                                              const float* __restrict__ W, int ldw,
                                              int m0, int n0, int K) {
  const int lane = threadIdx.x & 31;
  const int r16  = lane & 15;            // M index (A) / N index (B)
  const int kp   = (lane >> 4) << 1;     // K sub-offset 0 or 2
  const float* xrow = X + (m0 + r16) * ldx + kp;
  const float* wcol = W + kp * ldw + n0 + r16;
  v8f c = {0.f, 0.f, 0.f, 0.f, 0.f, 0.f, 0.f, 0.f};
#pragma unroll 4
  for (int k = 0; k < K; k += 4) {
    v2f a, b;
    a.x = xrow[k];
    a.y = xrow[k + 1];
    b.x = wcol[(k + 0) * ldw];
    b.y = wcol[(k + 1) * ldw];
    c = __builtin_amdgcn_wmma_f32_16x16x4_f32(
        /*neg_a=*/false, a, /*neg_b=*/false, b,
        /*c_mod=*/(short)0, c, /*reuse_a=*/false, /*reuse_b=*/false);
  }
  return c;
}

// ---------------------------------------------------------------------------
// K1: Xa = E @ mW1[0:128,:]   (z==0),   Xb = E @ mW1[128:256,:]  (z==1)
// grid (BN/64, D/16, 2), block 128 (4 waves, each one 16x16 tile along M).
// ---------------------------------------------------------------------------
__global__ void k_emb_proj(const float* __restrict__ E, const float* __restrict__ mW1,
                           float* __restrict__ Xa, float* __restrict__ Xb) {
  const int wave = threadIdx.x >> 5;
  const int m0 = (blockIdx.x * 4 + wave) * 16;
  const int n0 = blockIdx.y * 16;
  const float* W = mW1 + blockIdx.z * DDIM * DDIM;
  float* out = blockIdx.z ? Xb : Xa;
  v8f c = wmma_gemm_tile(E, DDIM, W, DDIM, m0, n0, DDIM);
  const int lane = threadIdx.x & 31;
  const int n = n0 + (lane & 15);
  const int mbase = m0 + ((lane >> 4) << 3);
#pragma unroll
  for (int r = 0; r < 8; ++r) out[(mbase + r) * DDIM + n] = c[r];
}

// ---------------------------------------------------------------------------
// K2: A[b,i,j] = exp(-0.5*||ci-cj||^2), rowsum[b,i] = sum_j A[b,i,j]
// One block per (b,i), 256 threads (one per j).
// ---------------------------------------------------------------------------
__global__ void k_adj(const float* __restrict__ coords, float* __restrict__ Amat,
                      float* __restrict__ rowsum) {
  __shared__ float sred[NPTS];
  const int bi = blockIdx.x;           // 0..1023
  const int b  = bi >> 8;
  const int j  = threadIdx.x;
  const float* ci = coords + bi * CDIM;
  const float* cj = coords + (b * NPTS + j) * CDIM;
  const float dx = ci[0] - cj[0];
  const float dy = ci[1] - cj[1];
  const float dz = ci[2] - cj[2];
  const float a = expf(-0.5f * (dx * dx + dy * dy + dz * dz));
  Amat[bi * NPTS + j] = a;
  sred[j] = a;
  __syncthreads();
#pragma unroll
  for (int s = NPTS / 2; s > 0; s >>= 1) {
    if (j < s) sred[j] += sred[j + s];
    __syncthreads();
  }
  if (j == 0) rowsum[bi] = sred[0];
}

// ---------------------------------------------------------------------------
// K3: R[b,i,d] = sum_j A[b,i,j] * relu(Xa[b,i,d] + Xb[b,j,d] + mb1[d])
// Block = 128 threads (one d each), handles 4 consecutive i of one batch.
// Xb streamed through LDS in 32-row tiles (16 KB), A tile in LDS too.
// ---------------------------------------------------------------------------
__global__ void k_msgred(const float* __restrict__ Xa, const float* __restrict__ Xb,
                         const float* __restrict__ mb1, const float* __restrict__ Amat,
                         float* __restrict__ R) {
  __shared__ float sXb[32][DDIM];
  __shared__ float sA[4][32];
  const int d   = threadIdx.x;         // 0..127
  const int bi0 = blockIdx.x * 4;      // 4 i's, same batch (4 | 256)
  const int b   = bi0 >> 8;
  const float mbv = mb1[d];
  float xa[4], acc[4] = {0.f, 0.f, 0.f, 0.f};
#pragma unroll
  for (int ii = 0; ii < 4; ++ii) xa[ii] = Xa[(bi0 + ii) * DDIM + d] + mbv;
  const float* XbB  = Xb + b * NPTS * DDIM;
  const float* Arow = Amat + bi0 * NPTS;
  const int ai = threadIdx.x >> 5;     // 0..3 (which i for the A-tile load)
  const int aj = threadIdx.x & 31;     // 0..31
  for (int jt = 0; jt < NPTS; jt += 32) {
    if (jt + 32 < NPTS)
      __builtin_prefetch(XbB + (jt + 32) * DDIM + d, 0, 1);   // global_prefetch_b8
#pragma unroll 8
    for (int k = 0; k < 32; ++k) sXb[k][d] = XbB[(jt + k) * DDIM + d];
    sA[ai][aj] = Arow[ai * NPTS + jt + aj];
    __syncthreads();
#pragma unroll 4
    for (int jj = 0; jj < 32; ++jj) {
      const float xbv = sXb[jj][d];
#pragma unroll
      for (int ii = 0; ii < 4; ++ii)
        acc[ii] += sA[ii][jj] * fmaxf(xa[ii] + xbv, 0.f);
    }
    __syncthreads();
  }
#pragma unroll
  for (int ii = 0; ii < 4; ++ii) R[(bi0 + ii) * DDIM + d] = acc[ii];
}

// ---------------------------------------------------------------------------
// K4: agg = R @ mW2 + rowsum[:,None] * mb2[None,:]
// ---------------------------------------------------------------------------
__global__ void k_agg(const float* __restrict__ R, const float* __restrict__ mW2,
                      const float* __restrict__ mb2, const float* __restrict__ rowsum,
                      float* __restrict__ agg) {
  const int wave = threadIdx.x >> 5;
  const int m0 = (blockIdx.x * 4 + wave) * 16;
  const int n0 = blockIdx.y * 16;
  v8f c = wmma_gemm_tile(R, DDIM, mW2, DDIM, m0, n0, DDIM);
  const int lane = threadIdx.x & 31;
  const int n = n0 + (lane & 15);
  const int mbase = m0 + ((lane >> 4) << 3);
  const float mb2n = mb2[n];
#pragma unroll
  for (int r = 0; r < 8; ++r) {
    const int m = mbase + r;
    agg[m * DDIM + n] = c[r] + rowsum[m] * mb2n;
  }
}

// ---------------------------------------------------------------------------
// K5: H = relu([agg | coords] @ uW1 + ub1)
//     = relu(agg @ uW1[0:128] + coords @ uW1[128:131] + ub1)
// The 3-column coords part is a cheap rank-3 epilogue (avoids K=131 tail).
// ---------------------------------------------------------------------------
__global__ void k_upd1(const float* __restrict__ agg, const float* __restrict__ coords,
                       const float* __restrict__ uW1, const float* __restrict__ ub1,
                       float* __restrict__ H) {
  const int wave = threadIdx.x >> 5;
  const int m0 = (blockIdx.x * 4 + wave) * 16;
  const int n0 = blockIdx.y * 16;
  v8f c = wmma_gemm_tile(agg, DDIM, uW1, DDIM, m0, n0, DDIM);
  const int lane = threadIdx.x & 31;
  const int n = n0 + (lane & 15);
  const int mbase = m0 + ((lane >> 4) << 3);
  const float w0 = uW1[(DDIM + 0) * DDIM + n];
  const float w1 = uW1[(DDIM + 1) * DDIM + n];
  const float w2 = uW1[(DDIM + 2) * DDIM + n];
  const float bn = ub1[n];
#pragma unroll
  for (int r = 0; r < 8; ++r) {
    const int m = mbase + r;
    const float v = c[r] + bn + coords[m * CDIM + 0] * w0
                              + coords[m * CDIM + 1] * w1
                              + coords[m * CDIM + 2] * w2;
    H[m * DDIM + n] = fmaxf(v, 0.f);
  }
}

// ---------------------------------------------------------------------------
// K6: out = local_emb + H @ uW2 + ub2   (residual fused into epilogue)
// ---------------------------------------------------------------------------
__global__ void k_upd2(const float* __restrict__ H, const float* __restrict__ uW2,
                       const float* __restrict__ ub2, const float* __restrict__ E,
                       float* __restrict__ out) {
  const int wave = threadIdx.x >> 5;
  const int m0 = (blockIdx.x * 4 + wave) * 16;
  const int n0 = blockIdx.y * 16;
  v8f c = wmma_gemm_tile(H, DDIM, uW2, DDIM, m0, n0, DDIM);
  const int lane = threadIdx.x & 31;
  const int n = n0 + (lane & 15);
  const int mbase = m0 + ((lane >> 4) << 3);
  const float bn = ub2[n];
#pragma unroll
  for (int r = 0; r < 8; ++r) {
    const int m = mbase + r;
    out[m * DDIM + n] = c[r] + bn + E[m * DDIM + n];
  }
}

// ---------------------------------------------------------------------------
extern "C" void kernel_launch(void* const* d_in, const int* in_sizes, int n_in,
                              void* d_out, int out_size, void* d_ws, size_t ws_size,
                              hipStream_t stream) {
  (void)in_sizes; (void)n_in; (void)out_size; (void)ws_size;
  const float* E      = (const float*)d_in[0];   // local_emb (B,N,D)
  const float* coords = (const float*)d_in[1];   // (B,N,3)
  const float* mW1    = (const float*)d_in[2];   // (2D,D)
  const float* mb1    = (const float*)d_in[3];   // (D,)
  const float* mW2    = (const float*)d_in[4];   // (D,D)
  const float* mb2    = (const float*)d_in[5];   // (D,)
  const float* uW1    = (const float*)d_in[6];   // (D+3,D)
  const float* ub1    = (const float*)d_in[7];   // (D,)
  const float* uW2    = (const float*)d_in[8];   // (D,D)
  const float* ub2    = (const float*)d_in[9];   // (D,)
  float* out = (float*)d_out;                    // (B,N,D)

  // Workspace carve-up (~3.6 MB of f32)
  float* ws     = (float*)d_ws;
  float* Xa     = ws;                              // BN*D
  float* Xb     = Xa   + BN * DDIM;                // BN*D
  float* Amat   = Xb   + BN * DDIM;                // B*N*N
  float* rowsum = Amat + BDIM * NPTS * NPTS;       // BN
  float* R      = rowsum + BN;                     // BN*D
  float* agg    = R    + BN * DDIM;                // BN*D
  float* H      = agg  + BN * DDIM;                // BN*D

  const dim3 gGemm(BN / 64, DDIM / 16, 1);         // 16 x 8 blocks, 4 waves each
  k_emb_proj<<<dim3(BN / 64, DDIM / 16, 2), 128, 0, stream>>>(E, mW1, Xa, Xb);
  k_adj     <<<BN, NPTS, 0, stream>>>(coords, Amat, rowsum);
  k_msgred  <<<BN / 4, DDIM, 0, stream>>>(Xa, Xb, mb1, Amat, R);
  k_agg     <<<gGemm, 128, 0, stream>>>(R, mW2, mb2, rowsum, agg);
  k_upd1    <<<gGemm, 128, 0, stream>>>(agg, coords, uW1, ub1, H);
  k_upd2    <<<gGemm, 128, 0, stream>>>(H, uW2, ub2, E, out);
}